// TemporalScaleMaximaAlignedLoss_1391569404168
// MI455X (gfx1250) — compile-verified
//
#include <hip/hip_runtime.h>

// ---------------------------------------------------------------------------
// TemporalScaleMaximaAlignedLoss for MI455X (gfx1250, wave32)
//   y_pred: [4096, 8192] f32, y_ori: [4096, 2048] f32, scale = 4
//   Memory-bound (160 MiB @ 23.3 TB/s ~= 7 us). Strategy:
//     - 1 block (256 thr = 8 wave32) per row
//     - y_ori row staged to LDS (8 KB) for neighbor access
//     - y_pred read as float4 (global_load_b128), perfectly coalesced:
//       one 16B load == one max-window of 4
//     - wave reduction via V_WMMA_F32_16X16X4_F32 with B = ones (exact f32)
//     - deterministic two-kernel reduction through d_ws (no float atomics)
// ---------------------------------------------------------------------------

typedef __attribute__((ext_vector_type(2))) float v2f;
typedef __attribute__((ext_vector_type(8))) float v8f;

#define ROWS  4096
#define T_LEN 2048
#define P_LEN 8192
#define BLK   256
#define WAVES (BLK / 32)

// Sum x across the 32 lanes of a wave using the f32 WMMA unit.
// A: 16x4 with VGPR0 = x (lanes hold A entries), VGPR1 = 0.
// B: 4x16 all-ones  =>  D(m,n) = rowsum_m (replicated over n).
// Per lane: sum of the 8 D VGPRs = half-column sum; xor-16 shuffle adds the
// other half => every lane holds the full 32-lane sum. Exact f32, no rounding
// tricks beyond reassociation. EXEC must be all 1s at the call site.
__device__ __forceinline__ float wave_sum_wmma(float x) {
    v2f a; a[0] = x;    a[1] = 0.0f;
    v2f b; b[0] = 1.0f; b[1] = 1.0f;
    v8f c = {};
    c = __builtin_amdgcn_wmma_f32_16x16x4_f32(
        /*neg_a=*/false, a, /*neg_b=*/false, b,
        /*c_mod=*/(short)0, c, /*reuse_a=*/false, /*reuse_b=*/false);
    float s = c[0] + c[1] + c[2] + c[3] + c[4] + c[5] + c[6] + c[7];
    s += __shfl_xor(s, 16, 32);
    return s;
}

__global__ __launch_bounds__(BLK)
void row_loss_kernel(const float* __restrict__ y_pred,
                     const float* __restrict__ y_ori,
                     float* __restrict__ row_loss,
                     float* __restrict__ row_valid) {
    __shared__ float s_ori[T_LEN];
    __shared__ float s_red[2][WAVES];

    const int row = blockIdx.x;
    const int t   = threadIdx.x;

    // Stage y_ori row into LDS: 2048 f32 = 512 float4, 2 per thread.
    const float4* oriv  = (const float4*)(y_ori + (size_t)row * T_LEN);
    float4*       sorив = (float4*)s_ori;
    sorив[t]       = oriv[t];
    sorив[t + BLK] = oriv[t + BLK];
    __syncthreads();

    const float4* predv = (const float4*)(y_pred + (size_t)row * P_LEN);

    float ssum = 0.0f;  // sum of masked squared errors
    float cnt  = 0.0f;  // number of maxima
#pragma unroll
    for (int it = 0; it < T_LEN / BLK; ++it) {
        const int j = t + it * BLK;              // window index 0..2047
        if (it + 1 < T_LEN / BLK) {
            // speculative streaming prefetch of next tile (global_prefetch_b8)
            __builtin_prefetch((const void*)&predv[j + BLK], 0, 0);
        }
        const float4 p = predv[j];               // one b128 load = one window
        const float wmax = fmaxf(fmaxf(p.x, p.y), fmaxf(p.z, p.w));

        const int jm = (j > 0)         ? j - 1 : 0;
        const int jp = (j < T_LEN - 1) ? j + 1 : T_LEN - 1;
        const float vj = s_ori[j];
        const float dL = vj - s_ori[jm];         // diff[:, :-1] at this site
        const float dR = s_ori[jp] - vj;         // diff[:, 1:]  at this site

        const bool interior = (j >= 1) && (j <= T_LEN - 2);
        const bool is_max   = interior && (dR * dL < 0.0f) && (dL > 0.0f);
        const float m = is_max ? 1.0f : 0.0f;    // branchless: EXEC stays full

        const float d = wmax - vj;
        ssum = fmaf(m, d * d, ssum);
        cnt += m;
    }

    // Wave32 reductions on the matrix unit (EXEC is all-1s here).
    const float wssum = wave_sum_wmma(ssum);
    const float wcnt  = wave_sum_wmma(cnt);

    const int wave = t >> 5;
    if ((t & 31) == 0) { s_red[0][wave] = wssum; s_red[1][wave] = wcnt; }
    __syncthreads();

    if (t == 0) {
        float rs = 0.0f, rc = 0.0f;
#pragma unroll
        for (int w = 0; w < WAVES; ++w) { rs += s_red[0][w]; rc += s_red[1][w]; }
        const bool  valid = rc > 0.0f;
        const float bl    = rs / fmaxf(rc, 1.0f);
        row_loss[row]  = valid ? bl : 0.0f;
        row_valid[row] = valid ? 1.0f : 0.0f;
    }
}

__global__ __launch_bounds__(1024)
void final_reduce_kernel(const float* __restrict__ row_loss,
                         const float* __restrict__ row_valid,
                         float* __restrict__ out) {
    __shared__ float sl[1024];
    __shared__ float sv[1024];
    const int t = threadIdx.x;

    float a = 0.0f, b = 0.0f;
#pragma unroll
    for (int i = t; i < ROWS; i += 1024) {   // fixed order: deterministic
        a += row_loss[i];
        b += row_valid[i];
    }
    sl[t] = a; sv[t] = b;
    __syncthreads();

#pragma unroll
    for (int s = 512; s > 0; s >>= 1) {      // fixed-shape tree: deterministic
        if (t < s) { sl[t] += sl[t + s]; sv[t] += sv[t + s]; }
        __syncthreads();
    }

    if (t == 0) {
        const float total = sl[0];
        const float nv    = sv[0];
        out[0] = (nv > 0.0f) ? total / fmaxf(nv, 1.0f) : 0.0f;
    }
}

extern "C" void kernel_launch(void* const* d_in, const int* in_sizes, int n_in,
                              void* d_out, int out_size, void* d_ws, size_t ws_size,
                              hipStream_t stream) {
    const float* y_pred = (const float*)d_in[0];  // [4096, 8192]
    const float* y_ori  = (const float*)d_in[1];  // [4096, 2048]
    float* ws        = (float*)d_ws;
    float* row_loss  = ws;                        // 4096 f32
    float* row_valid = ws + ROWS;                 // 4096 f32

    row_loss_kernel<<<ROWS, BLK, 0, stream>>>(y_pred, y_ori, row_loss, row_valid);
    final_reduce_kernel<<<1, 1024, 0, stream>>>(row_loss, row_valid, (float*)d_out);
}